// GCN_52561809768525
// MI455X (gfx1250) — compile-verified
//
#include <hip/hip_runtime.h>
#include <cstdint>
#include <cstddef>

// ---------------------------------------------------------------------------
// GAT (3-layer, PyG-style) for MI455X / gfx1250.
// GEMMs via V_WMMA_F32_16X16X4_F32 (full fp32, tensor path).
// Edge softmax/aggregation via 3 scatter passes (atomic max / add in L2).
// edge_index packed once to int2 (26 MB) so all edge passes stream L2-resident
// data instead of re-reading 51 MB of int64 indices from HBM per pass.
// ---------------------------------------------------------------------------

typedef __attribute__((ext_vector_type(2))) float v2f;
typedef __attribute__((ext_vector_type(8))) float v8f;

#define NEG_SLOPE 0.2f

// ---- float atomic max via ordered-int trick (init with -inf) --------------
__device__ __forceinline__ void atomic_max_f32(float* addr, float val) {
    if (val >= 0.0f) {
        atomicMax((int*)addr, __float_as_int(val));
    } else {
        atomicMin((unsigned int*)addr, __float_as_uint(val));
    }
}

// ---- generic fill ----------------------------------------------------------
__global__ void gat_fill(float* __restrict__ p, long n, float v) {
    long i = (long)blockIdx.x * blockDim.x + threadIdx.x;
    if (i < n) p[i] = v;
}

// ---- pack int64 edge_index [2,E] -> int2[(E+N)] with self-loops appended --
__global__ void gat_pack_edges(const long long* __restrict__ ei,
                               long Eraw, long Etot, int2* __restrict__ ep)
{
    long e = (long)blockIdx.x * blockDim.x + threadIdx.x;
    if (e >= Etot) return;
    int2 sd;
    if (e < Eraw) {
        sd.x = (int)ei[e];          // src
        sd.y = (int)ei[Eraw + e];   // dst
    } else {
        sd.x = sd.y = (int)(e - Eraw);
    }
    ep[e] = sd;
}

// ---- WMMA GEMM: H[N, NC] = X[N, K] @ W[K, NC], W zero-padded to 16 cols ---
// One wave (32 lanes) computes a 16x16 tile of H using V_WMMA_F32_16X16X4_F32.
// A (16x4 f32): lane l -> row = l&15, k-halves (l>>4)*2, 2 VGPRs (v2f).
// B (4x16 f32): lane l -> col = l&15, same k striping.
// C/D (16x16 f32): VGPR v, lane l -> row = v + 8*(l>>4), col = l&15.
__global__ __launch_bounds__(128) void gat_gemm_wmma(
    const float* __restrict__ X, const float* __restrict__ W,
    float* __restrict__ Hout, int Nrows, int K, int NC)
{
    __shared__ float sW[128 * 16];  // K <= 128, padded to 16 columns
    for (int i = threadIdx.x; i < K * 16; i += blockDim.x) {
        int k = i >> 4, n = i & 15;
        sW[i] = (n < NC) ? W[k * NC + n] : 0.0f;
    }
    __syncthreads();

    const int wave = threadIdx.x >> 5;      // wave32
    const int lane = threadIdx.x & 31;
    const int rowTile = (blockIdx.x * 4 + wave) * 16;
    if (rowTile >= Nrows) return;           // uniform per wave: EXEC stays full

    const int rl = lane & 15;
    const int kh = (lane >> 4) * 2;         // 0 or 2
    const int row = rowTile + rl;
    const float2* xr = (const float2*)(X + (size_t)row * K);

    v8f acc = {};
    for (int k0 = 0; k0 < K; k0 += 4) {
        float2 av = xr[(k0 + kh) >> 1];
        v2f a, b;
        a.x = av.x;
        a.y = av.y;
        b.x = sW[(k0 + kh)     * 16 + rl];
        b.y = sW[(k0 + kh + 1) * 16 + rl];
        acc = __builtin_amdgcn_wmma_f32_16x16x4_f32(
            false, a, false, b, (short)0, acc, false, false);
    }

    if (rl < NC) {
        const int rbase = rowTile + 8 * (lane >> 4);
#pragma unroll
        for (int v = 0; v < 8; ++v) {
            int r = rbase + v;
            if (r < Nrows) Hout[(size_t)r * NC + rl] = acc[v];
        }
    }
}

// ---- per-node attention terms: alpha_s[n,h] = sum_c h[n,h,c]*a_src[h,c] ---
template <int H, int C>
__global__ void gat_node_alpha(const float* __restrict__ h,
                               const float* __restrict__ a_src,
                               const float* __restrict__ a_dst,
                               float* __restrict__ asrc,
                               float* __restrict__ adst, int Nn)
{
    int n = blockIdx.x * blockDim.x + threadIdx.x;
    if (n >= Nn) return;
    constexpr int NC = H * C;
#pragma unroll
    for (int hh = 0; hh < H; ++hh) {
        float ss = 0.0f, sd = 0.0f;
#pragma unroll
        for (int c = 0; c < C; ++c) {
            float v = h[(size_t)n * NC + hh * C + c];
            ss += v * a_src[hh * C + c];
            sd += v * a_dst[hh * C + c];
        }
        asrc[n * H + hh] = ss;
        adst[n * H + hh] = sd;
    }
}

// ---- pass 1: segment max of leaky_relu(alpha_s[src]+alpha_d[dst]) ---------
template <int H>
__global__ void gat_edge_max(const int2* __restrict__ ep, long Etot,
                             const float* __restrict__ asrc,
                             const float* __restrict__ adst,
                             float* __restrict__ m)
{
    long e = (long)blockIdx.x * blockDim.x + threadIdx.x;
    if (e >= Etot) return;
    int2 sd = ep[e];
    const int s = sd.x, d = sd.y;
#pragma unroll
    for (int hh = 0; hh < H; ++hh) {
        float v = asrc[s * H + hh] + adst[d * H + hh];
        v = (v > 0.0f) ? v : NEG_SLOPE * v;
        atomic_max_f32(&m[d * H + hh], v);
    }
}

// ---- pass 2: ex = exp(e - m[dst]); denom[dst] += ex -----------------------
template <int H>
__global__ void gat_edge_expsum(const int2* __restrict__ ep, long Etot,
                                const float* __restrict__ asrc,
                                const float* __restrict__ adst,
                                const float* __restrict__ m,
                                float* __restrict__ den,
                                float* __restrict__ ex)
{
    long e = (long)blockIdx.x * blockDim.x + threadIdx.x;
    if (e >= Etot) return;
    int2 sd = ep[e];
    const int s = sd.x, d = sd.y;
#pragma unroll
    for (int hh = 0; hh < H; ++hh) {
        float v = asrc[s * H + hh] + adst[d * H + hh];
        v = (v > 0.0f) ? v : NEG_SLOPE * v;
        float xv = __expf(v - m[d * H + hh]);
        ex[(size_t)e * H + hh] = xv;
        atomicAdd(&den[d * H + hh], xv);
    }
}

// ---- pass 3: agg[dst] += h[src] * (ex / denom[dst]) -----------------------
template <int H, int C>
__global__ void gat_edge_agg(const int2* __restrict__ ep, long Etot,
                             const float* __restrict__ ex,
                             const float* __restrict__ den,
                             const float* __restrict__ hfeat,
                             float* __restrict__ agg)
{
    long e = (long)blockIdx.x * blockDim.x + threadIdx.x;
    if (e >= Etot) return;
    int2 sd = ep[e];
    const int s = sd.x, d = sd.y;
    constexpr int NC = H * C;
    constexpr int C4 = C / 4;          // C in {8,4}: always a multiple of 4
    const float4* hrow = (const float4*)(hfeat + (size_t)s * NC);
    float* arow = agg + (size_t)d * NC;
#pragma unroll
    for (int hh = 0; hh < H; ++hh) {
        float alpha = ex[(size_t)e * H + hh] / (den[d * H + hh] + 1e-16f);
#pragma unroll
        for (int c4 = 0; c4 < C4; ++c4) {
            float4 v = hrow[hh * C4 + c4];
            int base = hh * C + c4 * 4;
            atomicAdd(&arow[base + 0], v.x * alpha);
            atomicAdd(&arow[base + 1], v.y * alpha);
            atomicAdd(&arow[base + 2], v.z * alpha);
            atomicAdd(&arow[base + 3], v.w * alpha);
        }
    }
}

// ---- bias + optional ReLU, in place ---------------------------------------
__global__ void gat_bias_act(float* __restrict__ p, const float* __restrict__ b,
                             long total, int NC, int doRelu)
{
    long i = (long)blockIdx.x * blockDim.x + threadIdx.x;
    if (i >= total) return;
    int j = (int)(i % NC);
    float v = p[i] + b[j];
    if (doRelu) v = fmaxf(v, 0.0f);
    p[i] = v;
}

// ---- final classifier: out[N,2] = h3[N,4] @ Wc[4,2] + bc ------------------
__global__ void gat_classify(const float* __restrict__ hin,
                             const float* __restrict__ Wc,
                             const float* __restrict__ bc,
                             float* __restrict__ out, int Nn)
{
    int n = blockIdx.x * blockDim.x + threadIdx.x;
    if (n >= Nn) return;
    float4 h = ((const float4*)hin)[n];
    float o0 = bc[0] + h.x * Wc[0] + h.y * Wc[2] + h.z * Wc[4] + h.w * Wc[6];
    float o1 = bc[1] + h.x * Wc[1] + h.y * Wc[3] + h.z * Wc[5] + h.w * Wc[7];
    float2 o; o.x = o0; o.y = o1;
    ((float2*)out)[n] = o;
}

// ---------------------------------------------------------------------------
// Host-side layer driver (all launches on `stream`; graph-capture safe).
// ---------------------------------------------------------------------------
template <int H, int C>
static void run_gat_layer(const float* xin, int K,
                          const float* W, const float* a_s, const float* a_d,
                          const float* b, bool relu,
                          float* hbuf, float* agg, float* asrc, float* adst,
                          float* mbuf, float* dbuf, float* exbuf,
                          const int2* ep, long Etot, long Nn,
                          hipStream_t stream)
{
    constexpr int NC = H * C;
    const int TB = 256;

    // 1) h = xin @ W  (WMMA fp32)
    int tiles  = (int)((Nn + 15) / 16);
    int gblk   = (tiles + 3) / 4;                // 4 waves per 128-thread block
    gat_gemm_wmma<<<gblk, 128, 0, stream>>>(xin, W, hbuf, (int)Nn, K, NC);

    // 2) per-node attention terms
    int nb = (int)((Nn + TB - 1) / TB);
    gat_node_alpha<H, C><<<nb, TB, 0, stream>>>(hbuf, a_s, a_d, asrc, adst, (int)Nn);

    // 3) init segment buffers
    long nh = Nn * H;
    gat_fill<<<(unsigned)((nh + TB - 1) / TB), TB, 0, stream>>>(mbuf, nh, -__builtin_inff());
    gat_fill<<<(unsigned)((nh + TB - 1) / TB), TB, 0, stream>>>(dbuf, nh, 0.0f);
    long nagg = Nn * NC;
    gat_fill<<<(unsigned)((nagg + TB - 1) / TB), TB, 0, stream>>>(agg, nagg, 0.0f);

    // 4-6) edge passes (all over the packed, L2-resident int2 edge list)
    unsigned eb = (unsigned)((Etot + TB - 1) / TB);
    gat_edge_max<H><<<eb, TB, 0, stream>>>(ep, Etot, asrc, adst, mbuf);
    gat_edge_expsum<H><<<eb, TB, 0, stream>>>(ep, Etot, asrc, adst, mbuf, dbuf, exbuf);
    gat_edge_agg<H, C><<<eb, TB, 0, stream>>>(ep, Etot, exbuf, dbuf, hbuf, agg);

    // 7) bias + activation (in place in agg; agg becomes next layer's input)
    gat_bias_act<<<(unsigned)((nagg + TB - 1) / TB), TB, 0, stream>>>(
        agg, b, nagg, NC, relu ? 1 : 0);
}

extern "C" void kernel_launch(void* const* d_in, const int* in_sizes, int n_in,
                              void* d_out, int out_size, void* d_ws, size_t ws_size,
                              hipStream_t stream)
{
    const float*     x   = (const float*)d_in[0];
    const long long* ei  = (const long long*)d_in[1];   // int64 edge_index [2,E]
    const float* W1 = (const float*)d_in[2];
    const float* as1 = (const float*)d_in[3];
    const float* ad1 = (const float*)d_in[4];
    const float* b1 = (const float*)d_in[5];
    const float* W2 = (const float*)d_in[6];
    const float* as2 = (const float*)d_in[7];
    const float* ad2 = (const float*)d_in[8];
    const float* b2 = (const float*)d_in[9];
    const float* W3 = (const float*)d_in[10];
    const float* as3 = (const float*)d_in[11];
    const float* ad3 = (const float*)d_in[12];
    const float* b3 = (const float*)d_in[13];
    const float* Wc = (const float*)d_in[14];
    const float* bc = (const float*)d_in[15];

    const long Nn   = in_sizes[0] / 128;   // 100000
    const long Eraw = in_sizes[1] / 2;     // 3200000
    const long Etot = Eraw + Nn;           // + self loops

    // workspace layout (floats / ints): ~69 MB total
    float* ws    = (float*)d_ws;
    float* hbuf  = ws;                     // N*16
    float* abuf  = hbuf + Nn * 16;         // N*16  (agg / next-layer input)
    float* asrc  = abuf + Nn * 16;         // N*2
    float* adst  = asrc + Nn * 2;          // N*2
    float* mbuf  = adst + Nn * 2;          // N*2
    float* dbuf  = mbuf + Nn * 2;          // N*2
    float* exbuf = dbuf + Nn * 2;          // Etot*2 floats
    int2*  epbuf = (int2*)(exbuf + Etot * 2);  // Etot int2 (packed edges)

    const int TB = 256;

    // one-time: pack int64 edge_index (+ self-loops) into int2 stream
    gat_pack_edges<<<(unsigned)((Etot + TB - 1) / TB), TB, 0, stream>>>(
        ei, Eraw, Etot, epbuf);

    // layer 1: in=128, H=2, C=8, ReLU
    run_gat_layer<2, 8>(x, 128, W1, as1, ad1, b1, true,
                        hbuf, abuf, asrc, adst, mbuf, dbuf, exbuf,
                        epbuf, Etot, Nn, stream);
    // layer 2: in=16, H=2, C=4, ReLU (input = abuf; gemm reads it before refill)
    run_gat_layer<2, 4>(abuf, 16, W2, as2, ad2, b2, true,
                        hbuf, abuf, asrc, adst, mbuf, dbuf, exbuf,
                        epbuf, Etot, Nn, stream);
    // layer 3: in=8, H=1, C=4, no activation
    run_gat_layer<1, 4>(abuf, 8, W3, as3, ad3, b3, false,
                        hbuf, abuf, asrc, adst, mbuf, dbuf, exbuf,
                        epbuf, Etot, Nn, stream);

    // classifier
    gat_classify<<<(unsigned)((Nn + TB - 1) / TB), TB, 0, stream>>>(
        abuf, Wc, bc, (float*)d_out, (int)Nn);
}